// LatentTSDCD_25520695673178
// MI455X (gfx1250) — compile-verified
//
#include <hip/hip_runtime.h>
#include <hip/hip_bf16.h>
#include <math.h>

typedef __attribute__((ext_vector_type(2))) float v2f;
typedef __attribute__((ext_vector_type(8))) float v8f;

#define STD_CONST 0.0005f
#define B_  32
#define TAU 4
#define D_  6
#define DX  6144
#define K_  64
#define M_  384
#define NIN 1536
#define H_  128

// ---------------- kernel 0: w = exp(log_w), zero accumulators ----------------
__global__ void k_expw_init(const float* __restrict__ log_w,
                            float* __restrict__ w_exp,
                            float* __restrict__ accum) {
  int i = blockIdx.x * blockDim.x + threadIdx.x;
  if (i < D_ * DX * K_) w_exp[i] = expf(log_w[i]);
  if (i == 0) { accum[0] = 0.0f; accum[1] = 0.0f; }
}

// ---------------- kernel 1: encoder q_mu, z, q_mu_last ----------------
// grid = b*(tau+1)*d = 960 blocks of 64 threads (thread = k)
__global__ void __launch_bounds__(64)
k_encode(const float* __restrict__ x, const float* __restrict__ y,
         const float* __restrict__ w_exp, const float* __restrict__ eps_enc,
         float* __restrict__ z_ws, float* __restrict__ qml_ws) {
  int blk = blockIdx.x;
  int d = blk % D_;
  int bt = blk / D_;
  int t = bt % (TAU + 1);
  int b = bt / (TAU + 1);
  int k = threadIdx.x;

  const float* src = (t < TAU)
      ? (x + (size_t)((b * TAU + t) * D_ + d) * DX)
      : (y + (size_t)(b * D_ + d) * DX);
  const float* wp = w_exp + (size_t)d * DX * K_ + k;

  float acc = 0.0f;
  for (int dx = 0; dx < DX; ++dx)
    acc = fmaf(src[dx], wp[(size_t)dx * K_], acc);

  size_t zi = (size_t)(b * (TAU + 1) + t) * M_ + d * K_ + k;
  float ev = eps_enc[(size_t)((b * (TAU + 1) + t) * D_ + d) * K_ + k];
  z_ws[zi] = acc + STD_CONST * ev;
  if (t == TAU) qml_ws[b * M_ + d * K_ + k] = acc;
}

// ---------------- kernel 2: layer-0 grouped GEMM via f32 WMMA ----------------
// one block per m (384 blocks), 256 threads = 8 waves.
// out h1[m][32][128] = relu( inp[32x1536] @ W0[m][1536x128] + b0[m] )
#define AROW 388  // 384 + 4 pad (bank-conflict spread)
__global__ void __launch_bounds__(256)
k_gemm0(const float* __restrict__ mask_logits, const float* __restrict__ mask_u,
        const float* __restrict__ z_ws, const float* __restrict__ W0,
        const float* __restrict__ b0, float* __restrict__ h1_ws) {
  __shared__ float Ald[32 * AROW];
  __shared__ float sml[M_];

  int m = blockIdx.x;
  int tid = threadIdx.x;
  int lane = tid & 31;
  int wid = tid >> 5;
  int mt = wid >> 2;              // 0..1  (16-row A tile)
  int ntp = (wid & 3) * 2;        // two 16-col C tiles per wave
  int row = mt * 16 + (lane & 15);
  int kk = (lane >> 4) * 2;       // high-half lanes hold K+2/K+3
  int col0 = ntp * 16 + (lane & 15);
  int col1 = col0 + 16;

  v8f c0 = {};
  v8f c1 = {};

  for (int t = 0; t < TAU; ++t) {
    __syncthreads();
    for (int j = tid; j < M_; j += 256)
      sml[j] = mask_logits[(size_t)t * M_ * M_ + (size_t)j * M_ + m];
    __syncthreads();
    // stage inp[b][t*384+j] = sigmoid(logits + logistic(u)) * z_past[b][t][j]
    for (int idx = tid; idx < 32 * M_; idx += 256) {
      int b = idx / M_;
      int j = idx - b * M_;
      float u = mask_u[((size_t)(b * TAU + t) * M_ + j) * M_ + m];
      u = fminf(fmaxf(u, 1e-6f), 1.0f - 1e-6f);
      float logistic = logf(u) - log1pf(-u);
      float mv = 1.0f / (1.0f + expf(-(sml[j] + logistic)));
      Ald[b * AROW + j] = mv * z_ws[(size_t)(b * (TAU + 1) + t) * M_ + j];
    }
    __syncthreads();

    const float* w0p = W0 + (size_t)m * NIN * H_ + (size_t)t * M_ * H_;
    for (int k = 0; k < M_; k += 4) {
      v2f a, bf0, bf1;
      a.x = Ald[row * AROW + k + kk];
      a.y = Ald[row * AROW + k + kk + 1];
      const float* bp = w0p + (size_t)(k + kk) * H_;
      bf0.x = bp[col0];       bf0.y = bp[H_ + col0];
      bf1.x = bp[col1];       bf1.y = bp[H_ + col1];
      c0 = __builtin_amdgcn_wmma_f32_16x16x4_f32(false, a, false, bf0,
                                                 (short)0, c0, false, false);
      c1 = __builtin_amdgcn_wmma_f32_16x16x4_f32(false, a, false, bf1,
                                                 (short)0, c1, false, false);
    }
  }

  int half = lane >> 4;
#pragma unroll
  for (int v = 0; v < 8; ++v) {
    int r = mt * 16 + half * 8 + v;
    size_t o = (size_t)m * (32 * H_) + (size_t)r * H_;
    h1_ws[o + col0] = fmaxf(c0[v] + b0[m * H_ + col0], 0.0f);
    h1_ws[o + col1] = fmaxf(c1[v] + b0[m * H_ + col1], 0.0f);
  }
}

// ---------------- kernel 3: layer-1 WMMA GEMM + head + KL ----------------
#define HROW 132  // 128 + 4 pad
__global__ void __launch_bounds__(256)
k_gemm12(const float* __restrict__ h1_ws, const float* __restrict__ W1,
         const float* __restrict__ b1, const float* __restrict__ W2,
         const float* __restrict__ b2, const float* __restrict__ qml_ws,
         float* __restrict__ accum) {
  __shared__ float h1s[32 * HROW];
  __shared__ float h2s[32 * HROW];
  __shared__ float outs[64];

  int m = blockIdx.x;
  int tid = threadIdx.x;
  int lane = tid & 31;
  int wid = tid >> 5;
  int mt = wid >> 2;
  int ntp = (wid & 3) * 2;
  int row = mt * 16 + (lane & 15);
  int kk = (lane >> 4) * 2;
  int col0 = ntp * 16 + (lane & 15);
  int col1 = col0 + 16;

  for (int idx = tid; idx < 32 * H_; idx += 256)
    h1s[(idx >> 7) * HROW + (idx & 127)] = h1_ws[(size_t)m * (32 * H_) + idx];
  __syncthreads();

  v8f c0 = {};
  v8f c1 = {};
  const float* w1p = W1 + (size_t)m * H_ * H_;
  for (int k = 0; k < H_; k += 4) {
    v2f a, bf0, bf1;
    a.x = h1s[row * HROW + k + kk];
    a.y = h1s[row * HROW + k + kk + 1];
    const float* bp = w1p + (size_t)(k + kk) * H_;
    bf0.x = bp[col0];       bf0.y = bp[H_ + col0];
    bf1.x = bp[col1];       bf1.y = bp[H_ + col1];
    c0 = __builtin_amdgcn_wmma_f32_16x16x4_f32(false, a, false, bf0,
                                               (short)0, c0, false, false);
    c1 = __builtin_amdgcn_wmma_f32_16x16x4_f32(false, a, false, bf1,
                                               (short)0, c1, false, false);
  }

  int half = lane >> 4;
#pragma unroll
  for (int v = 0; v < 8; ++v) {
    int r = mt * 16 + half * 8 + v;
    h2s[r * HROW + col0] = fmaxf(c0[v] + b1[m * H_ + col0], 0.0f);
    h2s[r * HROW + col1] = fmaxf(c1[v] + b1[m * H_ + col1], 0.0f);
  }
  __syncthreads();

  // head: out[b][o] = h2[b,:] . W2[m,:,o] + b2[m,o]
  if (tid < 64) {
    int bi = tid >> 1;
    int o = tid & 1;
    float acc = b2[m * 2 + o];
    for (int h = 0; h < H_; ++h)
      acc = fmaf(h2s[bi * HROW + h], W2[((size_t)m * H_ + h) * 2 + o], acc);
    outs[tid] = acc;
  }
  __syncthreads();

  if (tid < 32) {
    float mu = outs[tid * 2];
    float s = 0.5f * expf(outs[tid * 2 + 1]);
    float qm = qml_ws[tid * M_ + m];
    const float qs = STD_CONST;
    float dmu = mu - qm;
    float kl = logf(qs / s) + (s * s + dmu * dmu) / (2.0f * qs * qs) - 0.5f;
    atomicAdd(&accum[0], kl);
  }
}

// ---------------- kernel 4: decoder + closed-form cross-batch recon ----------------
// grid = d*dx/256 = 144 blocks; each thread owns one (d,dx)
__global__ void __launch_bounds__(256)
k_decode(const float* __restrict__ y, const float* __restrict__ w_exp,
         const float* __restrict__ z_ws, float* __restrict__ accum) {
  __shared__ float zl[32 * K_];
  int tid = threadIdx.x;
  int pos = blockIdx.x * 256 + tid;
  int d = pos / DX;                 // uniform per block (256 | 6144)
  int dx = pos - d * DX;

  for (int idx = tid; idx < 32 * K_; idx += 256) {
    int b = idx >> 6;
    int k = idx & 63;
    zl[idx] = z_ws[(size_t)(b * (TAU + 1) + TAU) * M_ + d * K_ + k];
  }
  __syncthreads();

  float wv[K_];
  const float4* wp =
      reinterpret_cast<const float4*>(w_exp + ((size_t)d * DX + dx) * K_);
#pragma unroll
  for (int q = 0; q < K_ / 4; ++q) {
    float4 f = wp[q];
    wv[q * 4 + 0] = f.x; wv[q * 4 + 1] = f.y;
    wv[q * 4 + 2] = f.z; wv[q * 4 + 3] = f.w;
  }

  float Sp = 0.0f, Sp2 = 0.0f;
  for (int b = 0; b < 32; ++b) {
    float px = 0.0f;
#pragma unroll
    for (int k = 0; k < K_; ++k) px = fmaf(zl[b * K_ + k], wv[k], px);
    Sp += px;
    Sp2 = fmaf(px, px, Sp2);
  }
  float Sy = 0.0f, Sy2 = 0.0f;
  for (int b = 0; b < 32; ++b) {
    float yv = y[(size_t)(b * D_ + d) * DX + dx];
    Sy += yv;
    Sy2 = fmaf(yv, yv, Sy2);
  }
  // sum over (b1,b2) of (y[b2]-p[b1])^2 at this (d,dx)
  float sq = 32.0f * Sy2 - 2.0f * Sy * Sp + 32.0f * Sp2;
  atomicAdd(&accum[1], sq);
}

// ---------------- kernel 5: finalize ----------------
__global__ void k_final(const float* __restrict__ accum, float* __restrict__ out) {
  float kl = accum[0] * (1.0f / (32.0f * 384.0f));
  float msq = accum[1] / (32.0f * 32.0f * 6.0f * 6144.0f);
  const float ps = STD_CONST;
  float recons = -msq / (2.0f * ps * ps) - logf(ps)
                 - 0.5f * logf(2.0f * 3.14159265358979323846f);
  out[0] = recons - 1e-4f * kl;
  out[1] = recons;
  out[2] = kl;
}

extern "C" void kernel_launch(void* const* d_in, const int* in_sizes, int n_in,
                              void* d_out, int out_size, void* d_ws, size_t ws_size,
                              hipStream_t stream) {
  const float* x           = (const float*)d_in[0];
  const float* y           = (const float*)d_in[1];
  // d_in[2] = gt_z (unused)
  const float* log_w       = (const float*)d_in[3];
  const float* mask_logits = (const float*)d_in[4];
  const float* W0          = (const float*)d_in[5];
  const float* b0          = (const float*)d_in[6];
  const float* W1          = (const float*)d_in[7];
  const float* b1          = (const float*)d_in[8];
  const float* W2          = (const float*)d_in[9];
  const float* b2          = (const float*)d_in[10];
  const float* eps_enc     = (const float*)d_in[11];
  const float* mask_u      = (const float*)d_in[12];
  // d_in[13] = iteration (unused)

  float* ws    = (float*)d_ws;
  float* w_exp = ws;                         // 2,359,296
  float* z_ws  = w_exp + (size_t)D_ * DX * K_;   // 61,440
  float* qml   = z_ws + (size_t)B_ * (TAU + 1) * M_;  // 12,288
  float* h1_ws = qml + (size_t)B_ * M_;      // 1,572,864
  float* accum = h1_ws + (size_t)M_ * B_ * H_;  // 2

  int nexp = D_ * DX * K_;
  k_expw_init<<<(nexp + 255) / 256, 256, 0, stream>>>(log_w, w_exp, accum);
  k_encode<<<B_ * (TAU + 1) * D_, 64, 0, stream>>>(x, y, w_exp, eps_enc, z_ws, qml);
  k_gemm0<<<M_, 256, 0, stream>>>(mask_logits, mask_u, z_ws, W0, b0, h1_ws);
  k_gemm12<<<M_, 256, 0, stream>>>(h1_ws, W1, b1, W2, b2, qml, accum);
  k_decode<<<(D_ * DX) / 256, 256, 0, stream>>>(y, w_exp, z_ws, accum);
  k_final<<<1, 1, 0, stream>>>(accum, (float*)d_out);
}